// LCAConv_27195732918759
// MI455X (gfx1250) — compile-verified
//
#include <hip/hip_runtime.h>
#include <math.h>

typedef __attribute__((ext_vector_type(2))) float v2f;
typedef __attribute__((ext_vector_type(8))) float v8f;

#define N_NEUR 32
#define IN_C   3
#define KH     7
#define KW     7
#define HH     64
#define WW     64
#define BATCH  8
#define GK     13               // gram kernel size (2*KH-1)
#define GROW   418              // padded n-row stride in a dy-slice: 13*32 + 2
#define GSLICE (32 * GROW)      // 13376 dwords = 53504 B per dy-slice
#define THRESH 0.1f
#define INV_TAU (1.0f/100.0f)

// ---------------- per-sample standardization (mean/std over C,H,W; ddof=1) --
__global__ __launch_bounds__(256) void k_standardize(const float* __restrict__ x,
                                                     float* __restrict__ x2) {
  const int b = blockIdx.x;
  const int tid = threadIdx.x;
  const int n = IN_C * HH * WW;                 // 12288
  const float* xb = x + (size_t)b * n;
  float s = 0.f, ss = 0.f;
  for (int i = tid; i < n; i += 256) { float v = xb[i]; s += v; ss += v * v; }
  __shared__ float rs[256], rss[256];
  rs[tid] = s; rss[tid] = ss;
  __syncthreads();
  for (int off = 128; off > 0; off >>= 1) {
    if (tid < off) { rs[tid] += rs[tid + off]; rss[tid] += rss[tid + off]; }
    __syncthreads();
  }
  const float mean = rs[0] / (float)n;
  float var = (rss[0] - (float)n * mean * mean) / (float)(n - 1);
  var = fmaxf(var, 0.0f);
  const float inv = 1.0f / (sqrtf(var) + 1e-12f);
  float* ob = x2 + (size_t)b * n;
  for (int i = tid; i < n; i += 256) ob[i] = (xb[i] - mean) * inv;
}

// -------- Gram tensor, repacked padded: Gp2[dy][n][dx*32 + m], row 418 ------
__global__ __launch_bounds__(256) void k_gram(const float* __restrict__ D,
                                              float* __restrict__ Gp2) {
  const int idx = blockIdx.x * 256 + threadIdx.x;   // exactly 13*13*32*32
  const int m  = idx & 31;
  const int nn = (idx >> 5) & 31;
  const int dxy = idx >> 10;
  const int dx = dxy % GK, dy = dxy / GK;
  float acc = 0.f;
  for (int c = 0; c < IN_C; ++c)
    for (int ky = 0; ky < KH; ++ky) {
      const int iy = dy + ky - 6;
      if (iy < 0 || iy >= KH) continue;
      for (int kx = 0; kx < KW; ++kx) {
        const int ix = dx + kx - 6;
        if (ix < 0 || ix >= KW) continue;
        acc += D[((nn * IN_C + c) * KH + iy) * KW + ix] *
               D[((m  * IN_C + c) * KH + ky) * KW + kx];
      }
    }
  Gp2[(size_t)dy * GSLICE + nn * GROW + (dx << 5) + m] = acc;
}

// ---------------- input drive b = conv(x2, D, pad 3); also u = b/tau --------
__global__ __launch_bounds__(256) void k_bdrive(const float* __restrict__ x2,
                                                const float* __restrict__ D,
                                                float* __restrict__ bdrv,
                                                float* __restrict__ u) {
  __shared__ float sD[N_NEUR * IN_C * KH * KW];    // 4704 floats
  for (int i = threadIdx.x; i < N_NEUR * IN_C * KH * KW; i += 256) sD[i] = D[i];
  __syncthreads();
  const int idx = blockIdx.x * 256 + threadIdx.x;  // exactly 8*32*64*64
  const int xc = idx & 63;
  const int y  = (idx >> 6) & 63;
  const int nn = (idx >> 12) & 31;
  const int b  = idx >> 17;
  const float* xb = x2 + (size_t)b * IN_C * HH * WW;
  const float* wn = sD + nn * IN_C * KH * KW;
  float acc = 0.f;
  for (int c = 0; c < IN_C; ++c)
    for (int ky = 0; ky < KH; ++ky) {
      const int yy = y + ky - 3;
      if (yy < 0 || yy >= HH) continue;
      for (int kx = 0; kx < KW; ++kx) {
        const int xx = xc + kx - 3;
        if (xx < 0 || xx >= WW) continue;
        acc += xb[(c * HH + yy) * WW + xx] * wn[(c * KH + ky) * KW + kx];
      }
    }
  bdrv[idx] = acc;
  u[idx] = INV_TAU * acc;
}

// ---------------- elementwise soft threshold --------------------------------
__global__ __launch_bounds__(256) void k_soft(const float* __restrict__ u,
                                              float* __restrict__ a) {
  const int i = blockIdx.x * 256 + threadIdx.x;    // exact multiple
  const float v = u[i];
  const float t = fabsf(v) - THRESH;
  a[i] = (t > 0.f) ? copysignf(t, v) : 0.f;
}

// ---------------- fused lateral inhibition (WMMA f32) + membrane update -----
// block = (row y, batch); 4 waves x 16 pixels; each wave: 32 neurons x 16 px.
// Per dy: one staging phase (G dy-slice via b128 + transposed a-row), then
// 13 dx-taps x 16 WMMAs with zero barriers in between (208 WMMA / sync).
__global__ __launch_bounds__(128) void k_lca_update(const float* __restrict__ a,
                                                    const float* __restrict__ Gp2,
                                                    const float* __restrict__ bdrv,
                                                    float* __restrict__ u) {
  __shared__ __align__(16) float sG[GSLICE];   // 53504 B: [n][dx*32+m], pad 2
  __shared__ float sA[76 * 34];                // 10336 B: [x+halo][m], pad 2
  const int y    = blockIdx.x;                 // 0..63
  const int bt   = blockIdx.y;                 // 0..7
  const int tid  = threadIdx.x;                // 0..127
  const int lane = tid & 31;
  const int wave = tid >> 5;                   // 0..3
  const int nl   = lane & 15;
  const int kh   = lane >> 4;                  // 0 or 1

  v8f acc0 = {0.f,0.f,0.f,0.f,0.f,0.f,0.f,0.f};
  v8f acc1 = {0.f,0.f,0.f,0.f,0.f,0.f,0.f,0.f};

  const float* abase = a + (size_t)bt * N_NEUR * HH * WW;
  const int colbase = (wave << 4) + nl;

  for (int dy = 0; dy < GK; ++dy) {
    __syncthreads();                           // all reads of sA/sG done
    // ---- stage a-row, transposed: sA[j*34 + m] = a[bt][m][r][j-6] ----
    const int r = y + dy - 6;
    if (r >= 0 && r < HH) {
      const float* arow = abase + r * WW;
      for (int e = tid; e < 32 * 76; e += 128) {   // 2432 = 19*128, coalesced x
        const int m = e / 76;
        const int j = e - m * 76;
        const int xx = j - 6;
        sA[j * 34 + m] = (xx >= 0 && xx < WW) ? arow[m * HH * WW + xx] : 0.f;
      }
    } else {
      for (int e = tid; e < 32 * 76; e += 128) {
        const int m = e / 76;
        const int j = e - m * 76;
        sA[j * 34 + m] = 0.f;
      }
    }
    // ---- stage full G dy-slice with float4 (b128 global -> b128 LDS) ----
    {
      const float4* gsrc = (const float4*)(Gp2 + (size_t)dy * GSLICE);
      float4* gdst = (float4*)sG;
      for (int e = tid; e < GSLICE / 4; e += 128) gdst[e] = gsrc[e];
    }
    __syncthreads();
    // ---- 13 taps x 8 K-chunks x 2 neuron tiles of WMMA ----
    for (int dx = 0; dx < GK; ++dx) {
      const int bcol = (colbase + dx) * 34;    // sA column base for this tap
      const int gcol = (dx << 5);              // sG column base for this tap
      #pragma unroll
      for (int kc = 0; kc < 8; ++kc) {         // K = 32 channels, 4 per WMMA
        const int ko = (kc << 2) + (kh << 1);
        const v2f Bf = *(const v2f*)&sA[bcol + ko];            // B[k][pixel]
        const v2f A0 = *(const v2f*)&sG[nl * GROW + gcol + ko];// A[n][k], tile0
        acc0 = __builtin_amdgcn_wmma_f32_16x16x4_f32(
            false, A0, false, Bf, (short)0, acc0, false, false);
        const v2f A1 = *(const v2f*)&sG[(nl + 16) * GROW + gcol + ko]; // tile1
        acc1 = __builtin_amdgcn_wmma_f32_16x16x4_f32(
            false, A1, false, Bf, (short)0, acc1, false, false);
      }
    }
  }

  // epilogue: u += (1/tau)*(b - u - inhib + a)
  const int px = (wave << 4) + nl;
  #pragma unroll
  for (int rr = 0; rr < 8; ++rr) {
    const int n0 = rr + (kh << 3);                       // C layout: M=r+8*half
    const size_t i0 = (((size_t)bt * N_NEUR + n0) * HH + y) * WW + px;
    const float u0 = u[i0];
    u[i0] = u0 + INV_TAU * (bdrv[i0] - u0 - acc0[rr] + a[i0]);
    const size_t i1 = i0 + (size_t)16 * HH * WW;         // neuron tile 1
    const float u1 = u[i1];
    u[i1] = u1 + INV_TAU * (bdrv[i1] - u1 - acc1[rr] + a[i1]);
  }
}

// ---------------------------------------------------------------------------
extern "C" void kernel_launch(void* const* d_in, const int* in_sizes, int n_in,
                              void* d_out, int out_size, void* d_ws, size_t ws_size,
                              hipStream_t stream) {
  const float* x = (const float*)d_in[0];   // [8,3,1,64,64]
  const float* D = (const float*)d_in[1];   // [32,3,1,7,7] (unit-normalized)
  float* ws   = (float*)d_ws;
  float* x2   = ws;                         // 98304
  float* bdrv = x2 + 98304;                 // 1048576
  float* u    = bdrv + 1048576;             // 1048576
  float* a    = u + 1048576;                // 1048576
  float* Gp2  = a + 1048576;                // 13*13376 = 173888 (16B aligned)

  k_standardize<<<BATCH, 256, 0, stream>>>(x, x2);
  k_gram<<<(GK * GK * 32 * 32) / 256, 256, 0, stream>>>(D, Gp2);
  k_bdrive<<<(BATCH * N_NEUR * HH * WW) / 256, 256, 0, stream>>>(x2, D, bdrv, u);

  // u now = u_1 (scan step 0 collapses since u_0 = 0 -> a_0 = 0).
  // 8 full LCA iterations produce u_9; output = soft(u_9) (= scan's final a).
  for (int it = 0; it < 8; ++it) {
    k_soft<<<(BATCH * N_NEUR * HH * WW) / 256, 256, 0, stream>>>(u, a);
    k_lca_update<<<dim3(HH, BATCH), 128, 0, stream>>>(a, Gp2, bdrv, u);
  }
  k_soft<<<(BATCH * N_NEUR * HH * WW) / 256, 256, 0, stream>>>(u, (float*)d_out);
}